// Scale_44255343018823
// MI455X (gfx1250) — compile-verified
//
#include <hip/hip_runtime.h>

// y[n,c,h,w] = x[n,c,h,w] * gamma[c] + beta[c]
// x: (8, 128, 256, 256) fp32. Pure HBM-streaming op:
//   512 MB traffic / 23.3 TB/s  ->  ~22 us floor; 128 MFLOP is noise.
// Strategy: 128-bit non-temporal loads/stores (dataset 512 MB > 192 MB L2,
// so NT avoids L2 thrash), wave-uniform scalar loads for gamma/beta,
// 4 outstanding b128 loads per thread per iteration to cover HBM latency.

typedef float v4f __attribute__((ext_vector_type(4)));

#define HW        65536      // 256*256 elements per (n,c) plane
#define HW4       16384      // float4s per plane
#define THREADS   256        // 8 wave32s per block
#define VPT       4          // float4s per thread
#define BLKS_X    (HW4 / (THREADS * VPT))   // 16 blocks per plane
#define NPLANES   1024       // N*C = 8*128
#define NCH       128

__global__ __launch_bounds__(THREADS)
void Scale_44255343018823_kernel(const float* __restrict__ x,
                                 const float* __restrict__ gamma,
                                 const float* __restrict__ beta,
                                 float* __restrict__ y) {
    const unsigned plane = blockIdx.y;          // 0..1023  (n*128 + c)
    const unsigned c     = plane & (NCH - 1);   // channel; wave-uniform -> s_load
    const float g = gamma[c];
    const float b = beta[c];

    const size_t planeBase = (size_t)plane * HW;             // element offset
    const v4f* __restrict__ xin  = (const v4f*)(x + planeBase);
    v4f*       __restrict__ yout = (v4f*)(y + planeBase);

    const unsigned base = blockIdx.x * (THREADS * VPT) + threadIdx.x;

    // Issue all 4 b128 NT loads first (4 outstanding per thread), then FMA,
    // then 4 b128 NT stores. Compiler keeps these as load clauses.
    v4f v[VPT];
#pragma unroll
    for (int k = 0; k < VPT; ++k) {
        v[k] = __builtin_nontemporal_load(&xin[base + k * THREADS]);
    }
#pragma unroll
    for (int k = 0; k < VPT; ++k) {
        v[k].x = fmaf(v[k].x, g, b);
        v[k].y = fmaf(v[k].y, g, b);
        v[k].z = fmaf(v[k].z, g, b);
        v[k].w = fmaf(v[k].w, g, b);
    }
#pragma unroll
    for (int k = 0; k < VPT; ++k) {
        __builtin_nontemporal_store(v[k], &yout[base + k * THREADS]);
    }
}

extern "C" void kernel_launch(void* const* d_in, const int* in_sizes, int n_in,
                              void* d_out, int out_size, void* d_ws, size_t ws_size,
                              hipStream_t stream) {
    const float* x     = (const float*)d_in[0];
    const float* gamma = (const float*)d_in[1];
    const float* beta  = (const float*)d_in[2];
    float* y           = (float*)d_out;

    (void)in_sizes; (void)n_in; (void)out_size; (void)d_ws; (void)ws_size;

    dim3 grid(BLKS_X, NPLANES, 1);   // 16 x 1024 = 16384 blocks
    dim3 block(THREADS, 1, 1);
    Scale_44255343018823_kernel<<<grid, block, 0, stream>>>(x, gamma, beta, y);
}